// TsallisBisectLoss_12421045420955
// MI455X (gfx1250) — compile-verified
//
#include <hip/hip_runtime.h>
#include <hip/hip_bf16.h>
#include <stdint.h>

#define N_ROWS    4096
#define N_COLS    32000
#define NVEC4     (N_COLS / 4)          // 8000 float4 per row
#define R_PER_BLK 16
#define NBLK      (N_ROWS / R_PER_BLK)  // 256 blocks
#define NTHR      1024                  // 32 waves (wave32)
#define NWAVE     (NTHR / 32)
#define CAP       1024                  // candidate slots per row (expected ~450)

// ---- LDS byte layout (dynamic shared memory) ----
#define OFF_STAGE(b) ((b) * (N_COLS * 4))            // 2 x 128000 B staging
#define OFF_CAND     (2 * N_COLS * 4)                // 16 x 1024 x 4 = 64 KiB
#define OFF_WRED     (OFF_CAND + R_PER_BLK * CAP * 4)
#define OFF_WCNT     (OFF_WRED + NWAVE * 4)
#define OFF_BMAX     (OFF_WCNT + NWAVE * 4)
#define OFF_FLO      (OFF_BMAX + R_PER_BLK * 4)
#define OFF_NC       (OFF_FLO  + R_PER_BLK * 4)
#define SMEM_BYTES   (OFF_NC   + R_PER_BLK * 4)      // 321,984 B total

// (1/32000)^0.5 as the reference computes tau_hi = max - (1/d)^(alpha-1)
#define TAUHI_C 0.0055901699437494745f

#if defined(__AMDGCN__) && __has_builtin(__builtin_amdgcn_tensor_load_to_lds)
#define HAVE_TDM 1
typedef unsigned int u32x4 __attribute__((ext_vector_type(4)));
typedef int          i32x4 __attribute__((ext_vector_type(4)));
typedef int          i32x8 __attribute__((ext_vector_type(8)));

// Issue one TDM descriptor: copy one 32000-float row (128000 B) global -> LDS.
// D# per CDNA5 ISA ch.8: group0 {count,lds_addr,global_addr,type=2},
// group1 {data_size=4B, tensor_dim0=32000, tensor_dim1=1, tile_dim0=32000,
//         tile_dim1=1, tensor_dim0_stride=32000}. Groups 2/3 unused (<=2D).
__device__ __forceinline__ void tdm_load_row(const float* gptr, unsigned lds_off) {
  unsigned long long ga = (unsigned long long)(uintptr_t)gptr;
  u32x4 g0;
  g0[0] = 1u;                                   // count=1 (valid), no gather
  g0[1] = lds_off;                              // lds_addr (bytes)
  g0[2] = (unsigned)(ga & 0xFFFFFFFFull);       // global_addr[31:0]
  g0[3] = (unsigned)((ga >> 32) & 0x01FFFFFFull) | (2u << 30); // addr[56:32] | type=2
  i32x8 g1;
  g1[0] = 0x00020000;            // workgroup_mask=0, data_size=2 (4 bytes)
  g1[1] = (int)(0x7D00u << 16);  // tensor_dim0 low16 (32000) at bits 63:48
  g1[2] = (int)(1u << 16);       // tensor_dim0 hi16=0; tensor_dim1=1 (low16 at bits 95:80)
  g1[3] = (int)(0x7D00u << 16);  // tile_dim0 = 32000 at bits 127:112
  g1[4] = 1;                     // tile_dim1 = 1 (bits 143:128); tile_dim2 = 0
  g1[5] = N_COLS;                // tensor_dim0_stride low32
  g1[6] = 0;
  g1[7] = 0;
  i32x4 gz4 = {0, 0, 0, 0};
  i32x8 gz8 = {0, 0, 0, 0, 0, 0, 0, 0};
  // clang-23 / therock-10.0 arity: (g0, g1, g2, g3, extra, cpol)
  __builtin_amdgcn_tensor_load_to_lds(g0, g1, gz4, gz4, gz8, 0);
}
#endif

__device__ __forceinline__ float wave_sum(float v) {
#pragma unroll
  for (int off = 16; off; off >>= 1) v += __shfl_xor(v, off, 32);
  return v;
}
__device__ __forceinline__ float wave_fmax(float v) {
#pragma unroll
  for (int off = 16; off; off >>= 1) v = fmaxf(v, __shfl_xor(v, off, 32));
  return v;
}

__global__ __launch_bounds__(NTHR)
void tsallis_rows_kernel(const float* __restrict__ X, const int* __restrict__ tgt,
                         float* __restrict__ rowloss) {
  extern __shared__ char smem[];
  float*    wred  = (float*)(smem + OFF_WRED);
  unsigned* wcnt  = (unsigned*)(smem + OFF_WCNT);
  float*    bmaxA = (float*)(smem + OFF_BMAX);
  float*    floA  = (float*)(smem + OFF_FLO);
  unsigned* ncA   = (unsigned*)(smem + OFF_NC);
  float*    cand  = (float*)(smem + OFF_CAND);

  const int tid  = threadIdx.x;
  const int lane = tid & 31;
  const int wave = tid >> 5;
  const int rowBase = blockIdx.x * R_PER_BLK;

#ifdef HAVE_TDM
  if (wave == 0) tdm_load_row(X + (size_t)rowBase * N_COLS, OFF_STAGE(0));
#endif

  for (int rr = 0; rr < R_PER_BLK; ++rr) {
    const int buf = rr & 1;
    const float4* st4 = (const float4*)(smem + OFF_STAGE(buf));

#ifdef HAVE_TDM
    if (wave == 0) __builtin_amdgcn_s_wait_tensorcnt(0);  // row rr landed in LDS
    __syncthreads();
    if (wave == 0 && rr + 1 < R_PER_BLK)                   // prefetch next row
      tdm_load_row(X + (size_t)(rowBase + rr + 1) * N_COLS, OFF_STAGE((rr + 1) & 1));
#else
    {
      const float4* g4 = (const float4*)(X + (size_t)(rowBase + rr) * N_COLS);
      float4* s4w = (float4*)(smem + OFF_STAGE(buf));
#pragma unroll
      for (int i = 0; i < 8; ++i) {
        int j4 = tid + i * NTHR;
        if (j4 < NVEC4) s4w[j4] = g4[j4];
      }
      __syncthreads();
    }
#endif

    // ---- pass A: row max (on raw x; max(0.5x) == 0.5*max(x) exactly) ----
    float m = -3.402823466e38f;
#pragma unroll
    for (int i = 0; i < 8; ++i) {
      int j4 = tid + i * NTHR;
      if (j4 < NVEC4) {
        float4 v = st4[j4];
        m = fmaxf(m, fmaxf(fmaxf(v.x, v.y), fmaxf(v.z, v.w)));
      }
    }
    m = wave_fmax(m);
    if (lane == 0) wred[wave] = m;
    __syncthreads();
    float bx = wred[0];
#pragma unroll
    for (int w = 1; w < NWAVE; ++w) bx = fmaxf(bx, wred[w]);
    const float smax = 0.5f * bx;     // max of Xs = X*(alpha-1)
    const float th   = smax - 1.0f;   // initial tau_lo; only s > th ever matters
    __syncthreads();                  // wred reused below

    // ---- pass B: count candidates + f_lo partial ----
    unsigned k = 0; float floc = 0.0f;
#pragma unroll
    for (int i = 0; i < 8; ++i) {
      int j4 = tid + i * NTHR;
      if (j4 < NVEC4) {
        float4 v = st4[j4];
        float s;
        s = 0.5f * v.x; if (s > th) { float t = s - th; k++; floc += t * t; }
        s = 0.5f * v.y; if (s > th) { float t = s - th; k++; floc += t * t; }
        s = 0.5f * v.z; if (s > th) { float t = s - th; k++; floc += t * t; }
        s = 0.5f * v.w; if (s > th) { float t = s - th; k++; floc += t * t; }
      }
    }
    // deterministic wave inclusive prefix of k
    unsigned pk = k;
#pragma unroll
    for (int off = 1; off < 32; off <<= 1) {
      unsigned nbr = __shfl_up(pk, off, 32);
      if (lane >= off) pk += nbr;
    }
    unsigned wtot = __shfl(pk, 31, 32);
    float fw = wave_sum(floc);
    if (lane == 0) { wcnt[wave] = wtot; wred[wave] = fw; }
    __syncthreads();
    unsigned base = 0, nc = 0; float ftot = 0.0f;
#pragma unroll
    for (int w = 0; w < NWAVE; ++w) {   // fixed-order -> deterministic
      unsigned c = wcnt[w];
      if (w < wave) base += c;
      nc += c;
      ftot += wred[w];
    }
    unsigned pos = base + (pk - k);     // exclusive slot for this thread
    float* crow = cand + rr * CAP;
#pragma unroll
    for (int i = 0; i < 8; ++i) {       // compaction (deterministic layout)
      int j4 = tid + i * NTHR;
      if (j4 < NVEC4) {
        float4 v = st4[j4];
        float s;
        s = 0.5f * v.x; if (s > th) { if (pos < CAP) crow[pos] = s; pos++; }
        s = 0.5f * v.y; if (s > th) { if (pos < CAP) crow[pos] = s; pos++; }
        s = 0.5f * v.z; if (s > th) { if (pos < CAP) crow[pos] = s; pos++; }
        s = 0.5f * v.w; if (s > th) { if (pos < CAP) crow[pos] = s; pos++; }
      }
    }
    if (tid == 0) { bmaxA[rr] = smax; floA[rr] = ftot - 1.0f; ncA[rr] = nc; }
    __syncthreads();
  }

  // ---- phase 2+3: one wave per row, 50-step bisection over LDS candidates ----
  if (wave < R_PER_BLK) {
    const int rr  = wave;
    const int row = rowBase + rr;
    const float  smax = bmaxA[rr];
    const unsigned nc = ncA[rr];
    const float  f_lo = floA[rr];
    const float* crow = cand + rr * CAP;
    const float* gx   = X + (size_t)row * N_COLS;
    const bool   inl  = (nc <= CAP);    // fallback: stream from global (never for N(0,1))

    float tau_lo = smax - 1.0f;
    float tau_hi = smax - TAUHI_C;
    float dm     = tau_hi - tau_lo;
    float tau_m  = tau_lo;
    for (int it = 0; it < 50; ++it) {
      dm *= 0.5f;
      tau_m = tau_lo + dm;
      float sum = 0.0f;
      if (inl) {
        for (unsigned j = lane; j < nc; j += 32) {
          float t = crow[j] - tau_m;
          if (t > 0.0f) sum += t * t;
        }
      } else {
        for (int j = lane; j < N_COLS; j += 32) {
          float t = 0.5f * gx[j] - tau_m;
          if (t > 0.0f) sum += t * t;
        }
      }
      sum = wave_sum(sum);
      float f_m = sum - 1.0f;
      if (f_m * f_lo >= 0.0f) tau_lo = tau_m;
    }
    // final: s1 = sum p^1.5 = sum t^3 ; s2 = sum p*x = sum t^2 * (2c)
    float s1 = 0.0f, s2 = 0.0f;
    if (inl) {
      for (unsigned j = lane; j < nc; j += 32) {
        float c = crow[j];
        float t = c - tau_m;
        if (t > 0.0f) { float p = t * t; s1 += p * t; s2 += p * (c + c); }
      }
    } else {
      for (int j = lane; j < N_COLS; j += 32) {
        float xv = gx[j];
        float t = 0.5f * xv - tau_m;
        if (t > 0.0f) { float p = t * t; s1 += p * t; s2 += p * xv; }
      }
    }
    s1 = wave_sum(s1);
    s2 = wave_sum(s2);
    if (lane == 0) {
      int tg = tgt[row];
      float xt = gx[tg];
      rowloss[row] = (1.0f - s1) / 0.75f + s2 - xt;
    }
  }
}

__global__ __launch_bounds__(NTHR)
void tsallis_reduce_kernel(const float* __restrict__ rowloss, float* __restrict__ out) {
  __shared__ float red[NWAVE];
  const int tid = threadIdx.x, lane = tid & 31, wave = tid >> 5;
  float v = 0.0f;
#pragma unroll
  for (int i = 0; i < N_ROWS / NTHR; ++i) v += rowloss[tid + i * NTHR];
  v = wave_sum(v);
  if (lane == 0) red[wave] = v;
  __syncthreads();
  if (tid == 0) {
    float t = 0.0f;
#pragma unroll
    for (int w = 0; w < NWAVE; ++w) t += red[w];  // fixed order -> deterministic
    out[0] = t / (float)N_ROWS;
  }
}

extern "C" void kernel_launch(void* const* d_in, const int* in_sizes, int n_in,
                              void* d_out, int out_size, void* d_ws, size_t ws_size,
                              hipStream_t stream) {
  const float* inp = (const float*)d_in[0];
  const int*   tgt = (const int*)d_in[1];   // jax randint w/o x64 -> int32
  float* out = (float*)d_out;
  float* ws  = (float*)d_ws;                // 4096 per-row losses (16 KiB)

  tsallis_rows_kernel<<<NBLK, NTHR, SMEM_BYTES, stream>>>(inp, tgt, ws);
  tsallis_reduce_kernel<<<1, NTHR, 0, stream>>>(ws, out);
}